// Autoencoder_48326972015099
// MI455X (gfx1250) — compile-verified
//
#include <hip/hip_runtime.h>

typedef __attribute__((ext_vector_type(2))) float v2f;
typedef __attribute__((ext_vector_type(8))) float v8f;

#define NNODES 50000
#define NEDGES 400000
#define NGRAPHS 64

// ---------------------------------------------------------------- zero fill
__global__ void zero_f32(float* __restrict__ p, long long n) {
  long long i = (long long)blockIdx.x * blockDim.x + threadIdx.x;
  if (i < n) p[i] = 0.0f;
}

// ---------------------------------------------------------------- WMMA GEMM
// Out[M x (NL+NR)] = A[M x K] @ [Wl | Wr]  (row-major fp32, compile-time dims)
// One wave owns one 16-row tile of A and ALL N/16 output tiles (register
// blocking): A is read exactly once from HBM; the B panel is tiny and
// L1-resident.  Per K-step: one b64 A-load feeds NT independent
// V_WMMA_F32_16X16X4_F32 ops (distinct accumulators -> no RAW hazard).
// A 16x4 layout: lanes 0-15 -> M=0..15, K={k0,k0+1}; lanes 16-31 -> K={k0+2,k0+3}.
// B 4x16 mirrors it.  C/D: VGPR v, lane l -> M = v + (l>=16?8:0), N = l&15.
template <int K, int NL, int NR>
__global__ __launch_bounds__(256) void gemm_wmma_f32(const float* __restrict__ A,
                                                     const float* __restrict__ Wl,
                                                     const float* __restrict__ Wr,
                                                     float* __restrict__ Out,
                                                     int tilesM) {
  constexpr int N = NL + NR;
  constexpr int NT = N / 16;
  const int wave = (int)((blockIdx.x * blockDim.x + threadIdx.x) >> 5);
  const int lane = threadIdx.x & 31;
  if (wave >= tilesM) return;              // wave-uniform: EXEC all-1s for WMMA

  const int row0 = wave << 4;
  const int mn = lane & 15;                // M index for A, N index for B
  const int kk = (lane >> 4) << 1;         // 0 or 2

  const float* __restrict__ arow = A + (size_t)(row0 + mn) * K + kk;

  const float* bbase[NT];
#pragma unroll
  for (int t = 0; t < NT; ++t) {
    constexpr int c16 = 16;
    int col0 = t * c16;
    if (col0 < NL) bbase[t] = Wl + (size_t)kk * NL + col0 + mn;
    else           bbase[t] = Wr + (size_t)kk * NR + (col0 - NL) + mn;
  }

  v8f acc[NT] = {};

  for (int k0 = 0; k0 < K; k0 += 4) {
    v2f a;
    a.x = arow[k0];
    a.y = arow[k0 + 1];
#pragma unroll
    for (int t = 0; t < NT; ++t) {
      const int ldw = (t * 16 < NL) ? NL : NR;   // compile-time after unroll
      v2f b;
      b.x = bbase[t][(size_t)k0 * ldw];
      b.y = bbase[t][(size_t)(k0 + 1) * ldw];
      acc[t] = __builtin_amdgcn_wmma_f32_16x16x4_f32(
          /*neg_a=*/false, a, /*neg_b=*/false, b,
          /*c_mod=*/(short)0, acc[t], /*reuse_a=*/false, /*reuse_b=*/false);
    }
  }

  const int mofs = (lane >> 4) << 3;       // 0 or 8
#pragma unroll
  for (int t = 0; t < NT; ++t) {
#pragma unroll
    for (int v = 0; v < 8; ++v) {
      Out[(size_t)(row0 + v + mofs) * N + t * 16 + mn] = acc[t][v];
    }
  }
}

// ------------------------------------------------------- edge scatter (mean num.)
// agg[dst] += P[src][0:F]   (P rows of width LDP; F = 4 << LG4 features)
template <int LG4, int LDP, int F>
__global__ void edge_scatter4(const int* __restrict__ src, const int* __restrict__ dst,
                              const float* __restrict__ P, float* __restrict__ agg) {
  long long tid = (long long)blockIdx.x * blockDim.x + threadIdx.x;
  if (tid >= ((long long)NEDGES << LG4)) return;
  int e = (int)(tid >> LG4);
  int f4 = (int)(tid & ((1 << LG4) - 1)) << 2;
  int s = src[e], d = dst[e];
  float4 v = *(const float4*)(P + (size_t)s * LDP + f4);
  float* a = agg + (size_t)d * F + f4;
  atomicAdd(a + 0, v.x);
  atomicAdd(a + 1, v.y);
  atomicAdd(a + 2, v.z);
  atomicAdd(a + 3, v.w);
}

__global__ void edge_count(const int* __restrict__ dst, float* __restrict__ cnt) {
  int e = blockIdx.x * blockDim.x + threadIdx.x;
  if (e < NEDGES) atomicAdd(&cnt[dst[e]], 1.0f);
}

// ---------------------------------------------- finish: relu(agg/cnt + b + q)
template <int LOGF, int LDPQ, int QOFF>
__global__ void sage_finish(const float* __restrict__ agg, const float* __restrict__ cnt,
                            const float* __restrict__ PQ, const float* __restrict__ bias,
                            float* __restrict__ h) {
  constexpr int F = 1 << LOGF;
  long long tid = (long long)blockIdx.x * blockDim.x + threadIdx.x;
  if (tid >= ((long long)NNODES << LOGF)) return;
  int i = (int)(tid >> LOGF);
  int f = (int)(tid & (F - 1));
  float c = cnt[i]; c = c > 1.0f ? c : 1.0f;
  float v = agg[tid] / c + bias[f] + PQ[(size_t)i * LDPQ + QOFF + f];
  h[tid] = v > 0.0f ? v : 0.0f;
}

// ----------------------------------------------------------------- pooling
__global__ void pool_kernel(const float* __restrict__ h2, const int* __restrict__ batch,
                            float* __restrict__ gsum, float* __restrict__ gcnt) {
  long long tid = (long long)blockIdx.x * blockDim.x + threadIdx.x;
  if (tid >= (long long)NNODES * 32) return;
  int i = (int)(tid >> 5);
  int f = (int)(tid & 31);
  int b = batch[i];
  atomicAdd(&gsum[b * 32 + f], h2[tid]);
  if (f == 0) atomicAdd(&gcnt[b], 1.0f);
}

// --------------------------------------------------------------- head MLPs
__device__ __forceinline__ float leaky(float v) { return v > 0.0f ? v : 0.1f * v; }

__global__ void head_kernel(const float* __restrict__ gsum, const float* __restrict__ gcnt,
                            const float* __restrict__ Wl1, const float* __restrict__ bl1,
                            const float* __restrict__ Wl2, const float* __restrict__ bl2,
                            const float* __restrict__ Wd1, const float* __restrict__ bd1,
                            const float* __restrict__ Wd2, const float* __restrict__ bd2,
                            const float* __restrict__ Wd3, const float* __restrict__ bd3,
                            float* __restrict__ out) {
  int t = threadIdx.x;
  if (t >= NGRAPHS) return;
  float g[32], a1[32], enc[16], z1[32], z2[32];
  float c = gcnt[t]; c = c > 1.0f ? c : 1.0f;
#pragma unroll
  for (int j = 0; j < 32; ++j) g[j] = gsum[t * 32 + j] / c;
  for (int k = 0; k < 32; ++k) {
    float s = bl1[k];
    for (int j = 0; j < 32; ++j) s += g[j] * Wl1[j * 32 + k];
    a1[k] = s > 0.0f ? s : 0.0f;
  }
  for (int n = 0; n < 16; ++n) {
    float s = bl2[n];
    for (int k = 0; k < 32; ++k) s += a1[k] * Wl2[k * 16 + n];
    s = leaky(s);
    enc[n] = s;
    out[t * 16 + n] = s;                       // encoded [64,16]
  }
  for (int k = 0; k < 32; ++k) {
    float s = bd1[k];
    for (int n = 0; n < 16; ++n) s += enc[n] * Wd1[n * 32 + k];
    z1[k] = leaky(s);
  }
  for (int k = 0; k < 32; ++k) {
    float s = bd2[k];
    for (int j = 0; j < 32; ++j) s += z1[j] * Wd2[j * 32 + k];
    z2[k] = leaky(s);
  }
  for (int r = 0; r < 50; ++r) {
    float s = bd3[r];
    for (int k = 0; k < 32; ++k) s += z2[k] * Wd3[k * 50 + r];
    out[NGRAPHS * 16 + t * 50 + r] = s;        // reconstructed [64,50]
  }
}

// ---------------------------------------------------------------- launcher
extern "C" void kernel_launch(void* const* d_in, const int* in_sizes, int n_in,
                              void* d_out, int out_size, void* d_ws, size_t ws_size,
                              hipStream_t stream) {
  const float* x    = (const float*)d_in[0];
  const int*   src  = (const int*)d_in[1];
  const int*   dst  = src + NEDGES;
  const int*   bat  = (const int*)d_in[2];
  const float* W1l  = (const float*)d_in[3];
  const float* b1l  = (const float*)d_in[4];
  const float* W1r  = (const float*)d_in[5];
  const float* W2l  = (const float*)d_in[6];
  const float* b2l  = (const float*)d_in[7];
  const float* W2r  = (const float*)d_in[8];
  const float* Wl1  = (const float*)d_in[9];
  const float* bl1  = (const float*)d_in[10];
  const float* Wl2  = (const float*)d_in[11];
  const float* bl2  = (const float*)d_in[12];
  const float* Wd1  = (const float*)d_in[13];
  const float* bd1  = (const float*)d_in[14];
  const float* Wd2  = (const float*)d_in[15];
  const float* bd2  = (const float*)d_in[16];
  const float* Wd3  = (const float*)d_in[17];
  const float* bd3  = (const float*)d_in[18];
  float* out = (float*)d_out;

  // Workspace layout (floats)
  float* ws = (float*)d_ws;
  size_t oPQ1  = 0;                               // [50000, 256]  (p1 | q1)
  size_t oAGG1 = oPQ1  + (size_t)NNODES * 256;    // [50000, 128]
  size_t oCNT  = oAGG1 + (size_t)NNODES * 128;    // [50048]
  size_t oH1   = oCNT  + 50048;                   // [50000, 128]
  size_t oPQ2  = oH1   + (size_t)NNODES * 128;    // [50000, 64]   (p2 | q2)
  size_t oAGG2 = oPQ2  + (size_t)NNODES * 64;     // [50000, 32]
  size_t oH2   = oAGG2 + (size_t)NNODES * 32;     // [50000, 32]
  size_t oGS   = oH2   + (size_t)NNODES * 32;     // [64, 32]
  size_t oGC   = oGS   + NGRAPHS * 32;            // [64]

  const int B = 256;
  const int tilesM = NNODES / 16;                 // 3125 (exact)

  // ---- zero accumulators (every call: harness does not re-poison)
  {
    long long n1 = (long long)NNODES * 128 + 50048;        // AGG1 + CNT (contiguous)
    zero_f32<<<(int)((n1 + B - 1) / B), B, 0, stream>>>(ws + oAGG1, n1);
    long long n2 = (long long)NNODES * 32;                 // AGG2
    zero_f32<<<(int)((n2 + B - 1) / B), B, 0, stream>>>(ws + oAGG2, n2);
    long long n3 = NGRAPHS * 32 + NGRAPHS;                 // GSUM + GCNT (contiguous)
    zero_f32<<<(int)((n3 + B - 1) / B), B, 0, stream>>>(ws + oGS, n3);
  }

  // ---- layer 1: project BEFORE aggregating (linear commutes with mean)
  {
    int blocks = (tilesM * 32 + B - 1) / B;                // one wave per M-tile
    gemm_wmma_f32<512, 128, 128><<<blocks, B, 0, stream>>>(x, W1l, W1r,
                                                           ws + oPQ1, tilesM);
    long long work = (long long)NEDGES << 5;               // E * 128/4
    edge_scatter4<5, 256, 128><<<(int)((work + B - 1) / B), B, 0, stream>>>(
        src, dst, ws + oPQ1, ws + oAGG1);
    edge_count<<<(NEDGES + B - 1) / B, B, 0, stream>>>(dst, ws + oCNT);
    long long nf = (long long)NNODES * 128;
    sage_finish<7, 256, 128><<<(int)((nf + B - 1) / B), B, 0, stream>>>(
        ws + oAGG1, ws + oCNT, ws + oPQ1, b1l, ws + oH1);
  }

  // ---- layer 2
  {
    int blocks = (tilesM * 32 + B - 1) / B;
    gemm_wmma_f32<128, 32, 32><<<blocks, B, 0, stream>>>(ws + oH1, W2l, W2r,
                                                         ws + oPQ2, tilesM);
    long long work = (long long)NEDGES << 3;               // E * 32/4
    edge_scatter4<3, 64, 32><<<(int)((work + B - 1) / B), B, 0, stream>>>(
        src, dst, ws + oPQ2, ws + oAGG2);
    long long nf = (long long)NNODES * 32;
    sage_finish<5, 64, 32><<<(int)((nf + B - 1) / B), B, 0, stream>>>(
        ws + oAGG2, ws + oCNT, ws + oPQ2, b2l, ws + oH2);
  }

  // ---- pool + head
  {
    long long work = (long long)NNODES * 32;
    pool_kernel<<<(int)((work + B - 1) / B), B, 0, stream>>>(ws + oH2, bat,
                                                             ws + oGS, ws + oGC);
    head_kernel<<<1, 64, 0, stream>>>(ws + oGS, ws + oGC,
                                      Wl1, bl1, Wl2, bl2,
                                      Wd1, bd1, Wd2, bd2, Wd3, bd3, out);
  }
}